// MINE_76089640616151
// MI455X (gfx1250) — compile-verified
//
#include <hip/hip_runtime.h>
#include <stdint.h>

// CDNA5 / gfx1250 MINE kernel: fp32 WMMA (v_wmma_f32_16x16x4_f32) MLP pipeline
// with GLOBAL_LOAD_ASYNC_TO_LDS gather (ASYNCcnt) for the per-row feature fetch.

typedef float v2f __attribute__((ext_vector_type(2)));
typedef float v8f __attribute__((ext_vector_type(8)));

#define HD     300     // hidden width
#define DIN    256     // input features (2C)
#define ROWS   16      // rows per block (one wave32)
#define LSTR   300     // LDS row stride in floats (300 mod 64 = 44 -> conflict-free A reads)
#define BSIZE  20

__device__ __forceinline__ uint32_t hash_u32(uint32_t a) {
    a ^= a >> 16; a *= 0x7feb352du;
    a ^= a >> 15; a *= 0x846ca68bu;
    a ^= a >> 16;
    return a;
}

#define WMMA_F32(A, B, Cacc) \
    __builtin_amdgcn_wmma_f32_16x16x4_f32(false, (A), false, (B), (short)0, (Cacc), false, false)

__global__ __launch_bounds__(32)
void mine_mlp_kernel(const float* __restrict__ x, const int* __restrict__ ind,
                     const float* __restrict__ W1, const float* __restrict__ b1,
                     const float* __restrict__ W2, const float* __restrict__ b2,
                     const float* __restrict__ W3, const float* __restrict__ b3,
                     float* __restrict__ sumP, float* __restrict__ maxP,
                     float* __restrict__ seP, int n, int nblk)
{
    __shared__ float xs [ROWS * LSTR];   // gathered X (cols 0..255); reused as h2 (cols 0..299)
    __shared__ float h1s[ROWS * LSTR];   // relu(fc1) activations
    __shared__ float zs [ROWS];

    const int      t   = threadIdx.x;          // 0..31 (one wave)
    const uint32_t bid = blockIdx.x;
    const int      rb      = (int)(bid % (uint32_t)nblk);
    const uint32_t sv      = bid / (uint32_t)nblk;   // 0..39
    const int      variant = (int)(sv & 1u);         // 0=joint, 1=marginal
    const int      step    = (int)(sv >> 1);
    const int      row0    = rb * ROWS;
    const uint32_t nmask   = (uint32_t)n - 1u;       // n is a power of two (131072)

    // Per-step affine bijection standing in for jax.random.permutation.
    const uint32_t pa = hash_u32(0xBEEF0000u + (uint32_t)step) | 1u;
    const uint32_t pb = hash_u32(0xC0FFEE00u + (uint32_t)step);

    // ---- Phase 0: async gather, 2 lanes per row (lane = 2*row + half) ----
    // Each lane streams 128 contiguous floats (32 x b128) straight into LDS via
    // GLOBAL_LOAD_ASYNC_TO_LDS_B128 (no VGPR round trip, tracked by ASYNCcnt).
    {
        const int r    = t >> 1;                 // local row 0..15
        const int half = t & 1;                  // 0: cols 0..127, 1: cols 128..255
        const int rg   = row0 + r;
        uint32_t  src  = (uint32_t)ind[rg];
        if (variant && half) {                   // marginal: second half from permuted index
            src = (uint32_t)ind[(pa * (uint32_t)rg + pb) & nmask];
        }
        const float*   gsrc  = x + (size_t)src * DIN + half * 128;
        // Low 32 bits of a shared-memory pointer are the LDS byte offset
        // (aperture tag lives in addr[63:32]).
        const uint32_t lbase = (uint32_t)(uintptr_t)(xs + r * LSTR + half * 128);
#pragma unroll
        for (int q = 0; q < 32; ++q) {
            const uint64_t ga = (uint64_t)(uintptr_t)(gsrc + q * 4);
            const uint32_t la = lbase + (uint32_t)(q * 16);
            asm volatile("global_load_async_to_lds_b128 %0, %1, off"
                         :: "v"(la), "v"(ga) : "memory");
        }
        asm volatile("s_wait_asynccnt 0x0" ::: "memory");
    }
    __syncthreads();

    const int lane  = t;
    const int mrow  = lane & 15;            // A-fragment row this lane serves
    const int khalf = (lane >> 4) * 2;      // K offset: lanes 0-15 -> K+{0,1}, 16-31 -> K+{2,3}
    const int rhi   = 8 * (lane >> 4);      // C-fragment row offset for this half-wave

    // ---- Phase 1: h1 = relu(X @ W1^T + b1) ----
    // 9 unmasked tile-pairs (cols 0..287): one A load feeds two WMMAs.
    for (int p = 0; p < 9; ++p) {
        v8f acc0 = {}, acc1 = {};
        const int c0 = p * 32 + (lane & 15);
        const int c1 = c0 + 16;
        const float* w0 = W1 + (size_t)c0 * DIN;
        const float* w1 = W1 + (size_t)c1 * DIN;
#pragma unroll 4
        for (int k = 0; k < DIN; k += 4) {
            v2f a  = *(const v2f*)(xs + mrow * LSTR + k + khalf);
            v2f b0 = *(const v2f*)(w0 + k + khalf);
            v2f b1v = *(const v2f*)(w1 + k + khalf);
            acc0 = WMMA_F32(a, b0, acc0);
            acc1 = WMMA_F32(a, b1v, acc1);
        }
        const float bias0 = b1[c0];
        const float bias1 = b1[c1];
#pragma unroll
        for (int j = 0; j < 8; ++j) {
            float h0 = acc0[j] + bias0; h0 = h0 > 0.0f ? h0 : 0.0f;
            float h1 = acc1[j] + bias1; h1 = h1 > 0.0f ? h1 : 0.0f;
            h1s[(rhi + j) * LSTR + c0] = h0;
            h1s[(rhi + j) * LSTR + c1] = h1;
        }
    }
    // Masked tail tile (cols 288..303; lanes >= 300 zeroed branchlessly).
    {
        v8f acc = {};
        const int   ncol  = 288 + (lane & 15);
        const int   nc    = ncol < HD ? ncol : HD - 1;
        const float zmask = ncol < HD ? 1.0f : 0.0f;
        const float* wrow = W1 + (size_t)nc * DIN;
#pragma unroll 4
        for (int k = 0; k < DIN; k += 4) {
            v2f a = *(const v2f*)(xs + mrow * LSTR + k + khalf);
            const float2 wv = *(const float2*)(wrow + k + khalf);
            v2f b; b.x = wv.x * zmask; b.y = wv.y * zmask;
            acc = WMMA_F32(a, b, acc);
        }
        if (ncol < HD) {
            const float bias = b1[ncol];
#pragma unroll
            for (int j = 0; j < 8; ++j) {
                float hv = acc[j] + bias;
                hv = hv > 0.0f ? hv : 0.0f;
                h1s[(rhi + j) * LSTR + ncol] = hv;
            }
        }
    }
    __syncthreads();

    // ---- Phase 2: h2 = relu(dropout(h1 @ W2^T + b2)) ----
    const uint32_t dseed = hash_u32(0xD00D0000u + sv);
    for (int p = 0; p < 9; ++p) {
        v8f acc0 = {}, acc1 = {};
        const int c0 = p * 32 + (lane & 15);
        const int c1 = c0 + 16;
        const float* w0 = W2 + (size_t)c0 * HD;
        const float* w1 = W2 + (size_t)c1 * HD;
#pragma unroll 5
        for (int k = 0; k < HD; k += 4) {     // 300 % 4 == 0: exact coverage, no K padding
            v2f a  = *(const v2f*)(h1s + mrow * LSTR + k + khalf);
            v2f b0 = *(const v2f*)(w0 + k + khalf);
            v2f b1v = *(const v2f*)(w1 + k + khalf);
            acc0 = WMMA_F32(a, b0, acc0);
            acc1 = WMMA_F32(a, b1v, acc1);
        }
        const float bias0 = b2[c0];
        const float bias1 = b2[c1];
#pragma unroll
        for (int j = 0; j < 8; ++j) {
            const int rg = row0 + rhi + j;
            const uint32_t rh = dseed ^ ((uint32_t)rg * 0x9E3779B9u);
            float h0 = acc0[j] + bias0;
            float h1 = acc1[j] + bias1;
            const uint32_t bit0 = hash_u32(rh ^ ((uint32_t)c0 * 0x85EBCA6Bu)) & 1u;
            const uint32_t bit1 = hash_u32(rh ^ ((uint32_t)c1 * 0x85EBCA6Bu)) & 1u;
            h0 = bit0 ? 2.0f * h0 : 0.0f; h0 = h0 > 0.0f ? h0 : 0.0f;
            h1 = bit1 ? 2.0f * h1 : 0.0f; h1 = h1 > 0.0f ? h1 : 0.0f;
            xs[(rhi + j) * LSTR + c0] = h0;   // xs reused as h2 storage
            xs[(rhi + j) * LSTR + c1] = h1;
        }
    }
    // Masked tail tile (cols 288..303).
    {
        v8f acc = {};
        const int   ncol  = 288 + (lane & 15);
        const int   nc    = ncol < HD ? ncol : HD - 1;
        const float zmask = ncol < HD ? 1.0f : 0.0f;
        const float* wrow = W2 + (size_t)nc * HD;
#pragma unroll 5
        for (int k = 0; k < HD; k += 4) {
            v2f a = *(const v2f*)(h1s + mrow * LSTR + k + khalf);
            const float2 wv = *(const float2*)(wrow + k + khalf);
            v2f b; b.x = wv.x * zmask; b.y = wv.y * zmask;
            acc = WMMA_F32(a, b, acc);
        }
        if (ncol < HD) {
            const float bias = b2[ncol];
#pragma unroll
            for (int j = 0; j < 8; ++j) {
                const int rg = row0 + rhi + j;
                float hv = acc[j] + bias;
                const uint32_t bit = hash_u32(dseed ^ ((uint32_t)rg * 0x9E3779B9u)
                                                    ^ ((uint32_t)ncol * 0x85EBCA6Bu)) & 1u;
                hv = bit ? 2.0f * hv : 0.0f;
                hv = hv > 0.0f ? hv : 0.0f;
                xs[(rhi + j) * LSTR + ncol] = hv;
            }
        }
    }
    __syncthreads();

    // ---- Phase 3: z = h2 @ W3^T + b3, one row per lane ----
    if (t < ROWS) {
        float z = b3[0];
        const float* h2 = xs + t * LSTR;
        for (int k = 0; k < HD; k += 4) {
            z = fmaf(W3[k + 0], h2[k + 0], z);
            z = fmaf(W3[k + 1], h2[k + 1], z);
            z = fmaf(W3[k + 2], h2[k + 2], z);
            z = fmaf(W3[k + 3], h2[k + 3], z);
        }
        zs[t] = z;
    }
    __syncthreads();

    // ---- Phase 4: per-block partials (deterministic serial order) ----
    if (t == 0) {
        if (variant == 0) {
            float s = 0.0f;
            for (int i = 0; i < ROWS; ++i) s += zs[i];
            sumP[(size_t)step * nblk + rb] = s;
        } else {
            float m = zs[0];
            for (int i = 1; i < ROWS; ++i) m = fmaxf(m, zs[i]);
            float se = 0.0f;
            for (int i = 0; i < ROWS; ++i) se += __expf(zs[i] - m);
            maxP[(size_t)step * nblk + rb] = m;
            seP [(size_t)step * nblk + rb] = se;
        }
    }
}

__global__ __launch_bounds__(32)
void mine_reduce_kernel(const float* __restrict__ sumP, const float* __restrict__ maxP,
                        const float* __restrict__ seP, float* __restrict__ out,
                        int n, int nblk)
{
    __shared__ float mi[BSIZE];
    const int s = threadIdx.x;
    if (s < BSIZE) {
        const float* sp = sumP + (size_t)s * nblk;
        float tot = 0.0f;
        for (int b = 0; b < nblk; ++b) tot += sp[b];
        const float mean = tot / (float)n;

        const float* mp = maxP + (size_t)s * nblk;
        const float* ep = seP  + (size_t)s * nblk;
        float M = mp[0];
        for (int b = 1; b < nblk; ++b) M = fmaxf(M, mp[b]);
        float S = 0.0f;
        for (int b = 0; b < nblk; ++b) S += ep[b] * __expf(mp[b] - M);
        const float lse = M + __logf(S);
        mi[s] = mean - (lse - __logf((float)n));
    }
    __syncthreads();
    if (s == 0) {
        float acc = 0.0f;
        for (int i = 0; i < BSIZE; ++i) acc += mi[i];
        out[0] = acc / (float)BSIZE;
    }
}

extern "C" void kernel_launch(void* const* d_in, const int* in_sizes, int n_in,
                              void* d_out, int out_size, void* d_ws, size_t ws_size,
                              hipStream_t stream)
{
    (void)n_in; (void)out_size; (void)ws_size;
    const float* x   = (const float*)d_in[0];
    const int*   ind = (const int*)  d_in[1];   // harness delivers integer inputs as int32
    const float* W1  = (const float*)d_in[2];
    const float* b1  = (const float*)d_in[3];
    const float* W2  = (const float*)d_in[4];
    const float* b2  = (const float*)d_in[5];
    const float* W3  = (const float*)d_in[6];
    const float* b3  = (const float*)d_in[7];

    const int n    = in_sizes[1];       // 131072 gathered samples
    const int nblk = n / ROWS;          // 8192 row-blocks per (step, variant)

    float* sumP = (float*)d_ws;                       // [BSIZE][nblk]
    float* maxP = sumP + (size_t)BSIZE * nblk;        // [BSIZE][nblk]
    float* seP  = maxP + (size_t)BSIZE * nblk;        // [BSIZE][nblk]

    const unsigned grid = (unsigned)(nblk * BSIZE * 2);
    mine_mlp_kernel<<<grid, 32, 0, stream>>>(x, ind, W1, b1, W2, b2, W3, b3,
                                             sumP, maxP, seP, n, nblk);
    mine_reduce_kernel<<<1, 32, 0, stream>>>(sumP, maxP, seP, (float*)d_out, n, nblk);
}